// DAG_RNN_sw_12232066859245
// MI455X (gfx1250) — compile-verified
//
#include <hip/hip_runtime.h>

// CDNA5 / gfx1250, wave32. fp32 WMMA (V_WMMA_F32_16X16X4_F32).
//
// Diagonal-chain decomposition: cells with i+j = s form an independent chain.
// Block g owns 16 chains ({b,s=g} and {b,s=254-g}, equal length g+1).
// Fused recurrence + projection with a single A-stream pass per step feeding
// two accumulators:
//   carry  acc1 = x(t) + h(t-1) @ Whh        (pre-activation)
//   step:  h(t) = relu(acc1); publish; barrier;
//          one LDS pass over h(t):  acc2 = ol(t) + h(t) @ Wyh   -> y(t)
//                                   acc1 = x(t+1) + h(t) @ Whh  -> next
// Weights staged in LDS once; B fragments live in VGPRs; H uses an
// A-fragment-contiguous permuted LDS layout.

typedef float v2f __attribute__((ext_vector_type(2)));
typedef float v4f __attribute__((ext_vector_type(4)));
typedef float v8f __attribute__((ext_vector_type(8)));

constexpr int Cdim = 128, Hdim = 128, Wdim = 128;

// LDS layout (floats)
constexpr int W_STRIDE = 136;  // weight row stride (conflict-free staging reads)
constexpr int R_STRIDE = 148;  // Hp row (M) stride: 16B-aligned, odd 4-bank phase
constexpr int H_HALF   = 72;   // Hp half-wave stride: 16B-aligned
constexpr int OFF_WHH  = 0;
constexpr int OFF_WYH  = OFF_WHH + Cdim * W_STRIDE;
constexpr int OFF_H0   = OFF_WYH + Cdim * W_STRIDE;
constexpr int OFF_H1   = OFF_H0  + 16 * R_STRIDE;
constexpr int LDS_FLOATS = OFF_H1 + 16 * R_STRIDE;    // ~155 KB

extern __shared__ float lds[];

__device__ __forceinline__ v2f mkv2(float a, float b) { v2f r; r.x = a; r.y = b; return r; }

__global__ __launch_bounds__(256)
void dag_rnn_sw_fused(const float* __restrict__ x,
                      const float* __restrict__ out_last,
                      const float* __restrict__ whh,
                      const float* __restrict__ wyh,
                      float* __restrict__ out)
{
    const int tid  = threadIdx.x;
    const int lane = tid & 31;
    const int wv   = tid >> 5;        // wave 0..7 owns N-columns [16*wv, 16*wv+16)
    const int g    = blockIdx.x;      // chain group 0..127
    const int Lg   = g + 1;           // chain length

    // ---- stage weights into LDS (coalesced, once) ----
    for (int idx = tid; idx < Cdim * Cdim; idx += 256) {
        const int k = idx >> 7, n = idx & 127;
        lds[OFF_WHH + k * W_STRIDE + n] = whh[idx];
        lds[OFF_WYH + k * W_STRIDE + n] = wyh[idx];
    }
    __syncthreads();

    const int n0 = wv * 16;
    const int lr = lane & 15;         // N within tile / A-fragment row M
    const int lh = lane >> 4;         // half-wave selector

    // ---- hoist loop-invariant B fragments into VGPRs (128 VGPRs) ----
    v2f bh[32], by[32];
#pragma unroll
    for (int kk = 0; kk < 32; ++kk) {
        const int ka = kk * 4 + 2 * lh;
        bh[kk].x = lds[OFF_WHH + ka * W_STRIDE + n0 + lr];
        bh[kk].y = lds[OFF_WHH + (ka + 1) * W_STRIDE + n0 + lr];
        by[kk].x = lds[OFF_WYH + ka * W_STRIDE + n0 + lr];
        by[kk].y = lds[OFF_WYH + (ka + 1) * W_STRIDE + n0 + lr];
    }

    // ---- per-lane flat cell addresses (shared by x / out_last / out) ----
    // every chain advances by -(W-1) = -127 per step.
    int addr[8];
#pragma unroll
    for (int v = 0; v < 8; ++v) {
        const int rm = v + 8 * lh;
        int bm, im, jm;
        if (rm < 8) { bm = rm;     im = g;   jm = 0;       }
        else        { bm = rm - 8; im = 127; jm = 127 - g; }
        addr[v] = ((bm * Cdim + n0 + lr) * Hdim + im) * Wdim + jm;
    }

    // Permuted H layout: element (M,K) at M*R + ((K>>1)&1)*HH + (K>>2)*2 + (K&1)
    const int abase = lr * R_STRIDE + lh * H_HALF;      // A-stream read base
    const int colK  = n0 + lr;                           // D column = next A's K
    const int koff  = ((colK >> 1) & 1) * H_HALF + (colK >> 2) * 2 + (colK & 1);

    // ---- prime: acc1 = x(0) (= pre-activation 0), xv = x(1), ov = ol(0) ----
    v8f acc1;
#pragma unroll
    for (int v = 0; v < 8; ++v) acc1[v] = x[addr[v]];
    float xv[8], ov[8];
#pragma unroll
    for (int v = 0; v < 8; ++v) ov[v] = out_last[addr[v]];
    if (Lg > 1) {
#pragma unroll
        for (int v = 0; v < 8; ++v) xv[v] = x[addr[v] - (Wdim - 1)];
    }

    int hp = 0;
    for (int t = 0; t < Lg; ++t) {
        const int hbuf = hp ? OFF_H1 : OFF_H0;

        // h(t) = relu(acc1); publish into ping-pong buffer (permuted layout)
#pragma unroll
        for (int v = 0; v < 8; ++v)
            lds[hbuf + (v + 8 * lh) * R_STRIDE + koff] = fmaxf(acc1[v], 0.0f);

        // rotate prefetched C-inits
        v8f acc2;
#pragma unroll
        for (int v = 0; v < 8; ++v) acc2[v] = ov[v];       // ol(t)
#pragma unroll
        for (int v = 0; v < 8; ++v) acc1[v] = xv[v];       // x(t+1)

        // issue next prefetches (uniform guards; offsets fold into imm)
        if (t + 2 < Lg) {
#pragma unroll
            for (int v = 0; v < 8; ++v) xv[v] = x[addr[v] - 2 * (Wdim - 1)];
        }
        if (t + 1 < Lg) {
#pragma unroll
            for (int v = 0; v < 8; ++v) ov[v] = out_last[addr[v] - (Wdim - 1)];
        }

        __syncthreads();               // single barrier per step

        // ---- merged pass over h(t): y(t) and next pre-activation ----
#pragma unroll
        for (int q = 0; q < 8; ++q) {
            const v4f a0 = *(const v4f*)&lds[hbuf + abase + q * 8];
            const v4f a1 = *(const v4f*)&lds[hbuf + abase + q * 8 + 4];
            v2f a;
            a = mkv2(a0.x, a0.y);
            acc2 = __builtin_amdgcn_wmma_f32_16x16x4_f32(
                false, a, false, by[4 * q + 0], (short)0, acc2, false, false);
            acc1 = __builtin_amdgcn_wmma_f32_16x16x4_f32(
                false, a, false, bh[4 * q + 0], (short)0, acc1, false, false);
            a = mkv2(a0.z, a0.w);
            acc2 = __builtin_amdgcn_wmma_f32_16x16x4_f32(
                false, a, false, by[4 * q + 1], (short)0, acc2, false, false);
            acc1 = __builtin_amdgcn_wmma_f32_16x16x4_f32(
                false, a, false, bh[4 * q + 1], (short)0, acc1, false, false);
            a = mkv2(a1.x, a1.y);
            acc2 = __builtin_amdgcn_wmma_f32_16x16x4_f32(
                false, a, false, by[4 * q + 2], (short)0, acc2, false, false);
            acc1 = __builtin_amdgcn_wmma_f32_16x16x4_f32(
                false, a, false, bh[4 * q + 2], (short)0, acc1, false, false);
            a = mkv2(a1.z, a1.w);
            acc2 = __builtin_amdgcn_wmma_f32_16x16x4_f32(
                false, a, false, by[4 * q + 3], (short)0, acc2, false, false);
            acc1 = __builtin_amdgcn_wmma_f32_16x16x4_f32(
                false, a, false, bh[4 * q + 3], (short)0, acc1, false, false);
        }

        // scatter y(t), advance chain addresses along the diagonal
#pragma unroll
        for (int v = 0; v < 8; ++v) out[addr[v]] = acc2[v];
#pragma unroll
        for (int v = 0; v < 8; ++v) addr[v] -= (Wdim - 1);

        hp ^= 1;
    }
}

extern "C" void kernel_launch(void* const* d_in, const int* in_sizes, int n_in,
                              void* d_out, int out_size, void* d_ws, size_t ws_size,
                              hipStream_t stream) {
    (void)in_sizes; (void)n_in; (void)out_size; (void)d_ws; (void)ws_size;
    const float* x        = (const float*)d_in[0];
    const float* out_last = (const float*)d_in[1];
    const float* whh      = (const float*)d_in[2];
    const float* wyh      = (const float*)d_in[3];
    float* out            = (float*)d_out;

    const size_t lds_bytes = (size_t)LDS_FLOATS * sizeof(float);
    dag_rnn_sw_fused<<<dim3(128), dim3(256), lds_bytes, stream>>>(
        x, out_last, whh, wyh, out);
}